// GraphCell_73426760893048
// MI455X (gfx1250) — compile-verified
//
#include <hip/hip_runtime.h>
#include <math.h>

// ---------------- problem constants (match reference) ----------------
#define N_NODES  50000
#define N_EDGES  800000
#define D        128
#define N_GRAPHS 64

typedef __attribute__((ext_vector_type(2))) float v2f;
typedef __attribute__((ext_vector_type(8))) float v8f;

// ---------------------------------------------------------------------
// fill kernels
// ---------------------------------------------------------------------
__global__ void fill_f32(float* __restrict__ p, float v, int n) {
    int i = blockIdx.x * blockDim.x + threadIdx.x;
    if (i < n) p[i] = v;
}

// ---------------------------------------------------------------------
// degree: deg[dst] += 1  (deg buffer pre-zeroed)
// ---------------------------------------------------------------------
__global__ void deg_scatter(const int* __restrict__ dst, float* __restrict__ deg) {
    int e = blockIdx.x * blockDim.x + threadIdx.x;
    if (e < N_EDGES) atomicAdd(&deg[dst[e]], 1.0f);
}

// dinv[i] = rsqrt(deg[i] + 1)   (in place)
__global__ void deg_rsqrt(float* __restrict__ deg) {
    int i = blockIdx.x * blockDim.x + threadIdx.x;
    if (i < N_NODES) deg[i] = rsqrtf(deg[i] + 1.0f);
}

// ---------------------------------------------------------------------
// WMMA GEMM: H[N,128] = X[N,128] @ W[128,128]   (fp32, V_WMMA_F32_16X16X4_F32)
// block = 128 threads (4 waves); each wave owns one 16-row tile, 8 col tiles.
// W is staged in LDS in a K-pair-interleaved layout:
//     Wp[(k/2)*256 + n*2 + (k&1)]  ==  W[k][n]
// so a lane's B fragment (K = ka, ka+1 at column n) is one aligned b64 load.
// ---------------------------------------------------------------------
__global__ __launch_bounds__(128)
void gemm_wmma_f32(const float* __restrict__ X, const float* __restrict__ W,
                   float* __restrict__ H, int nRows) {
    __shared__ float Wp[D * D];   // 64 KB, interleaved layout

    // cooperative load of W into LDS (coalesced float4 reads, interleaved writes)
    const float4* __restrict__ Wv = (const float4*)W;
    #pragma unroll 4
    for (int i = threadIdx.x; i < (D * D) / 4; i += 128) {
        float4 w = Wv[i];
        const int k    = i >> 5;            // 32 float4 per W row
        const int n0   = (i & 31) * 4;
        const int base = (k >> 1) * 256 + (k & 1);
        Wp[base + (n0 + 0) * 2] = w.x;
        Wp[base + (n0 + 1) * 2] = w.y;
        Wp[base + (n0 + 2) * 2] = w.z;
        Wp[base + (n0 + 3) * 2] = w.w;
    }
    __syncthreads();

    const int wave    = threadIdx.x >> 5;              // 0..3
    const int lane    = threadIdx.x & 31;
    const int half    = lane >> 4;                     // 0 | 1
    const int lr      = lane & 15;                     // 0..15
    const int rowBase = (blockIdx.x * 4 + wave) * 16;  // wave-uniform

    if (rowBase < nRows) {                             // uniform per wave -> EXEC all ones
        v8f acc[8];
        #pragma unroll
        for (int c = 0; c < 8; ++c) acc[c] = (v8f){0.f,0.f,0.f,0.f,0.f,0.f,0.f,0.f};

        // per-lane A row pointer, viewed as K-pairs (8B aligned: row stride 512B)
        const v2f* __restrict__ xr2 = (const v2f*)(X + (size_t)(rowBase + lr) * D);
        const v2f* __restrict__ Wp2 = (const v2f*)Wp;  // [pair*128 + n]

        for (int k0 = 0; k0 < D; k0 += 4) {
            const int ka = k0 + 2 * half;              // lane's K pair (even)
            // A fragment (16x4 f32): lane = row lr, VGPRs = {X[ka], X[ka+1]}
            v2f a = xr2[ka >> 1];                      // global_load_b64
            const int pbase = (ka >> 1) * 128 + lr;
            #pragma unroll
            for (int c = 0; c < 8; ++c) {
                // B fragment (4x16 f32): lane = col, VGPRs = {W[ka][n], W[ka+1][n]}
                v2f b = Wp2[pbase + c * 16];           // ds_load_b64, even-aligned
                acc[c] = __builtin_amdgcn_wmma_f32_16x16x4_f32(
                    /*neg_a=*/false, a, /*neg_b=*/false, b,
                    /*c_mod=*/(short)0, acc[c],
                    /*reuse_a=*/false, /*reuse_b=*/false);
            }
        }

        // D tile store: VGPR r <-> row r + 8*half, col = c*16 + lr
        #pragma unroll
        for (int c = 0; c < 8; ++c) {
            #pragma unroll
            for (int r = 0; r < 8; ++r) {
                H[(size_t)(rowBase + r + 8 * half) * D + c * 16 + lr] = acc[c][r];
            }
        }
    }
}

// ---------------------------------------------------------------------
// edge scatter: agg[dst] += h[src] * (dinv[src]*dinv[dst])
// 32 lanes per edge, float4 gather, 4 fp32 atomics per lane
// ---------------------------------------------------------------------
__global__ __launch_bounds__(256)
void edge_scatter(const float* __restrict__ h, const int* __restrict__ src,
                  const int* __restrict__ dst, const float* __restrict__ dinv,
                  float* __restrict__ agg) {
    const int tid  = blockIdx.x * blockDim.x + threadIdx.x;
    const int e    = tid >> 5;
    const int lane = tid & 31;
    if (e >= N_EDGES) return;

    const int s = src[e];
    const int d = dst[e];
    const float norm = dinv[s] * dinv[d];

    const float4 v = ((const float4*)(h + (size_t)s * D))[lane];
    float* __restrict__ ad = agg + (size_t)d * D + lane * 4;
    atomicAdd(ad + 0, v.x * norm);
    atomicAdd(ad + 1, v.y * norm);
    atomicAdd(ad + 2, v.z * norm);
    atomicAdd(ad + 3, v.w * norm);
}

// ---------------------------------------------------------------------
// combine: out = agg + dinv^2 * h + bias, optional ReLU
// ---------------------------------------------------------------------
__global__ void combine(const float* __restrict__ agg, const float* __restrict__ h,
                        const float* __restrict__ dinv, const float* __restrict__ bias,
                        float* __restrict__ out, int relu) {
    int i = blockIdx.x * blockDim.x + threadIdx.x;
    if (i >= N_NODES * D) return;
    const int n = i >> 7;       // / D
    const int f = i & (D - 1);  // % D
    const float di = dinv[n];
    float v = agg[i] + di * di * h[i] + bias[f];
    if (relu) v = fmaxf(v, 0.0f);
    out[i] = v;
}

// ---------------------------------------------------------------------
// segment_max pool: out[batch[n], f] = max(...)   (CAS float max)
// ---------------------------------------------------------------------
__device__ __forceinline__ void atomicMaxF(float* addr, float val) {
    unsigned int* ua = (unsigned int*)addr;
    unsigned int old = __float_as_uint(*addr);
    while (__uint_as_float(old) < val) {
        unsigned int assumed = old;
        old = atomicCAS(ua, assumed, __float_as_uint(val));
        if (old == assumed) break;
    }
}

__global__ void pool_max(const float* __restrict__ x, const int* __restrict__ batch,
                         float* __restrict__ out) {
    int i = blockIdx.x * blockDim.x + threadIdx.x;
    if (i >= N_NODES * D) return;
    const int n = i >> 7;
    const int f = i & (D - 1);
    atomicMaxF(&out[batch[n] * D + f], x[i]);
}

// ---------------------------------------------------------------------
// host-side orchestration
// ---------------------------------------------------------------------
extern "C" void kernel_launch(void* const* d_in, const int* in_sizes, int n_in,
                              void* d_out, int out_size, void* d_ws, size_t ws_size,
                              hipStream_t stream) {
    (void)in_sizes; (void)n_in; (void)out_size; (void)ws_size;

    const float* x    = (const float*)d_in[0];
    const float* W1   = (const float*)d_in[1];
    const float* b1   = (const float*)d_in[2];
    const float* W2   = (const float*)d_in[3];
    const float* b2   = (const float*)d_in[4];
    const float* W3   = (const float*)d_in[5];
    const float* b3   = (const float*)d_in[6];
    const int*   eidx = (const int*)d_in[7];
    const int*   srcI = eidx;
    const int*   dstI = eidx + N_EDGES;
    const int*   batch = (const int*)d_in[8];
    float* out = (float*)d_out;

    // workspace layout (float offsets, all 16B-aligned)
    const size_t ND   = (size_t)N_NODES * D;       // 6 400 000
    float* ws    = (float*)d_ws;
    float* dinv  = ws;                              // 50 000 (padded to 50 176)
    float* bufH  = ws + 50176;                      // h = x @ W
    float* bufA  = bufH + ND;                       // agg
    float* bufX1 = bufA + ND;                       // ping
    float* bufX2 = bufX1 + ND;                      // pong

    const int nd    = (int)ND;                      // 6 400 000, fits int
    const int TB    = 256;
    const int gFill = (nd + TB - 1) / TB;
    const int gEdge = (N_EDGES + TB - 1) / TB;
    const int gScat = ((N_EDGES * 32) + TB - 1) / TB;
    const int gNode = (N_NODES + TB - 1) / TB;
    const int gGemm = ((N_NODES + 63) / 64);        // 4 row-tiles (64 rows) per block

    // --- degree normalization ---
    fill_f32<<<gNode, TB, 0, stream>>>(dinv, 0.0f, N_NODES);
    deg_scatter<<<gEdge, TB, 0, stream>>>(dstI, dinv);
    deg_rsqrt<<<gNode, TB, 0, stream>>>(dinv);

    // --- layer 1: in = x, out = bufX1, ReLU ---
    gemm_wmma_f32<<<gGemm, 128, 0, stream>>>(x, W1, bufH, N_NODES);
    fill_f32<<<gFill, TB, 0, stream>>>(bufA, 0.0f, nd);
    edge_scatter<<<gScat, TB, 0, stream>>>(bufH, srcI, dstI, dinv, bufA);
    combine<<<gFill, TB, 0, stream>>>(bufA, bufH, dinv, b1, bufX1, 1);

    // --- layer 2: in = bufX1, out = bufX2, ReLU ---
    gemm_wmma_f32<<<gGemm, 128, 0, stream>>>(bufX1, W2, bufH, N_NODES);
    fill_f32<<<gFill, TB, 0, stream>>>(bufA, 0.0f, nd);
    edge_scatter<<<gScat, TB, 0, stream>>>(bufH, srcI, dstI, dinv, bufA);
    combine<<<gFill, TB, 0, stream>>>(bufA, bufH, dinv, b2, bufX2, 1);

    // --- layer 3: in = bufX2, out = bufX1, no ReLU ---
    gemm_wmma_f32<<<gGemm, 128, 0, stream>>>(bufX2, W3, bufH, N_NODES);
    fill_f32<<<gFill, TB, 0, stream>>>(bufA, 0.0f, nd);
    edge_scatter<<<gScat, TB, 0, stream>>>(bufH, srcI, dstI, dinv, bufA);
    combine<<<gFill, TB, 0, stream>>>(bufA, bufH, dinv, b3, bufX1, 0);

    // --- global max pool ---
    fill_f32<<<(N_GRAPHS * D + TB - 1) / TB, TB, 0, stream>>>(out, -INFINITY, N_GRAPHS * D);
    pool_max<<<gFill, TB, 0, stream>>>(bufX1, batch, out);
}